// DiffJPEG_40587440947390
// MI455X (gfx1250) — compile-verified
//
#include <hip/hip_runtime.h>
#include <math.h>

typedef float v2f __attribute__((ext_vector_type(2)));
typedef float v4f __attribute__((ext_vector_type(4)));
typedef float v8f __attribute__((ext_vector_type(8)));

#define INV_SQRT2 0.70710678118654752f
#define QFACTOR 0.4f                 // (200 - 2*80)/100

// DCT-II basis M[x][u] = cos((2x+1)*u*pi/16), row-major [x*8+u]
__constant__ float c_M[64] = {
  1.0f,  0.98078528f,  0.92387953f,  0.83146961f,  0.70710678f,  0.55557023f,  0.38268343f,  0.19509032f,
  1.0f,  0.83146961f,  0.38268343f, -0.19509032f, -0.70710678f, -0.98078528f, -0.92387953f, -0.55557023f,
  1.0f,  0.55557023f, -0.38268343f, -0.98078528f, -0.70710678f,  0.19509032f,  0.92387953f,  0.83146961f,
  1.0f,  0.19509032f, -0.92387953f, -0.55557023f,  0.70710678f,  0.83146961f, -0.38268343f, -0.98078528f,
  1.0f, -0.19509032f, -0.92387953f,  0.55557023f,  0.70710678f, -0.83146961f, -0.38268343f,  0.98078528f,
  1.0f, -0.55557023f, -0.38268343f,  0.98078528f, -0.70710678f, -0.19509032f,  0.92387953f, -0.83146961f,
  1.0f, -0.83146961f,  0.38268343f,  0.19509032f, -0.70710678f,  0.98078528f, -0.92387953f,  0.55557023f,
  1.0f, -0.98078528f,  0.92387953f, -0.83146961f,  0.70710678f, -0.55557023f,  0.38268343f, -0.19509032f};
// Transpose: MT[u][x] = M[x][u]
__constant__ float c_MT[64] = {
  1.0f, 1.0f, 1.0f, 1.0f, 1.0f, 1.0f, 1.0f, 1.0f,
  0.98078528f,  0.83146961f,  0.55557023f,  0.19509032f, -0.19509032f, -0.55557023f, -0.83146961f, -0.98078528f,
  0.92387953f,  0.38268343f, -0.38268343f, -0.92387953f, -0.92387953f, -0.38268343f,  0.38268343f,  0.92387953f,
  0.83146961f, -0.19509032f, -0.98078528f, -0.55557023f,  0.55557023f,  0.98078528f,  0.19509032f, -0.83146961f,
  0.70710678f, -0.70710678f, -0.70710678f,  0.70710678f,  0.70710678f, -0.70710678f, -0.70710678f,  0.70710678f,
  0.55557023f, -0.98078528f,  0.19509032f,  0.83146961f, -0.83146961f, -0.19509032f,  0.98078528f, -0.55557023f,
  0.38268343f, -0.92387953f,  0.92387953f, -0.38268343f, -0.38268343f,  0.92387953f, -0.92387953f,  0.38268343f,
  0.19509032f, -0.55557023f,  0.83146961f, -0.98078528f,  0.98078528f, -0.83146961f,  0.55557023f, -0.19509032f};

// Standard libjpeg luma table row-major [v*8+u]; this IS the transpose of the
// reference's Y_TABLE (=std.T), giving per-lane rows Q[u][v] for fixed v.
__constant__ float c_Ystd[64] = {
  16,11,10,16, 24, 40, 51, 61,
  12,12,14,19, 26, 58, 60, 55,
  14,13,16,24, 40, 57, 69, 56,
  14,17,22,29, 51, 87, 80, 62,
  18,22,37,56, 68,109,103, 77,
  24,35,55,64, 81,104,113, 92,
  49,64,78,87,103,121,120,101,
  72,92,95,98,112,100,103, 99};
// Chroma table (symmetric => transpose-invariant)
__constant__ float c_Cstd[64] = {
  17,18,24,47,99,99,99,99,
  18,21,26,66,99,99,99,99,
  24,26,56,99,99,99,99,99,
  47,66,99,99,99,99,99,99,
  99,99,99,99,99,99,99,99,
  99,99,99,99,99,99,99,99,
  99,99,99,99,99,99,99,99,
  99,99,99,99,99,99,99,99};

__device__ __forceinline__ v8f wmma4(v2f a, v2f b, v8f acc) {
  return __builtin_amdgcn_wmma_f32_16x16x4_f32(false, a, false, b,
                                               (short)0, acc, false, false);
}

// D = F * R, F = basis A-fragments in registers, R row-major 16x16 in LDS
__device__ __forceinline__ v8f mm_regA(const v2f (&F)[4], const float* R, int lane) {
  const int n = lane & 15;
  const int half = (lane >> 4) << 1;
  v8f acc = {};
#pragma unroll
  for (int k = 0; k < 4; ++k) {
    const int kk = 4 * k + half;
    v2f b;
    b.x = R[kk * 16 + n];
    b.y = R[(kk + 1) * 16 + n];
    acc = wmma4(F[k], b, acc);
  }
  return acc;
}

// D = L * F, L row-major 16x16 in LDS, F = basis B-fragments in registers
__device__ __forceinline__ v8f mm_regB(const float* L, const v2f (&F)[4], int lane) {
  const int m = lane & 15;
  const int half = (lane >> 4) << 1;
  v8f acc = {};
#pragma unroll
  for (int k = 0; k < 4; ++k) {
    const int kk = 4 * k + half;            // even -> 8B aligned
    v2f a = *(const v2f*)&L[m * 16 + kk];
    acc = wmma4(a, F[k], acc);
  }
  return acc;
}

// C/D 16x16 f32 layout: VGPR r holds row (r + 8*(lane>=16)), column = lane&15
__device__ __forceinline__ void storeFrag(float* Out, v8f d, int lane) {
  const int n  = lane & 15;
  const int mo = (lane & 16) ? 8 : 0;
#pragma unroll
  for (int r = 0; r < 8; ++r) Out[(r + mo) * 16 + n] = d[r];
}

// DCT -> quant -> diff-round -> dequant -> IDCT on one 16x16 plane P.
__device__ __forceinline__ void process_plane(float* P,
                                              const float (&qv)[8], const float (&iq)[8],
                                              const v2f (&FA)[4], const v2f (&FB)[4],
                                              float* T1, float* T2, int lane, float av) {
  const int n  = lane & 15;
  const int mo = (lane & 16) ? 8 : 0;

  v8f t = mm_regA(FB, P, lane);            // T1 = G^T * P
  storeFrag(T1, t, lane);
  __syncthreads();

  v8f d = mm_regB(T1, FB, lane);           // D = T1 * G  (raw DCT)
#pragma unroll
  for (int r = 0; r < 8; ++r) {            // u == r, v == lane&7
    const float au = (r == 0) ? INV_SQRT2 : 1.0f;    // compile-time
    const float e  = d[r] * (0.25f * au) * av;       // SCALE * dct
    const float q  = e * iq[r];                      // quantize
    const float rr = rintf(q);
    const float dq = q - rr;
    T2[(r + mo) * 16 + n] = (rr + dq * dq * dq) * (qv[r] * au) * av;
  }
  __syncthreads();

  t = mm_regA(FA, T2, lane);               // T1 = G * T2
  storeFrag(T1, t, lane);
  __syncthreads();

  d = mm_regB(T1, FA, lane);               // Z = T1 * G^T
#pragma unroll
  for (int r = 0; r < 8; ++r) P[(r + mo) * 16 + n] = 0.25f * d[r] + 128.0f;
  __syncthreads();
}

__global__ __launch_bounds__(32)
void diffjpeg_tile(const float* __restrict__ x, float* __restrict__ out) {
  __shared__ alignas(16) float Y[256];
  __shared__ alignas(16) float C[256];
  __shared__ alignas(16) float T1[256];
  __shared__ alignas(16) float T2[256];

  const int lane = threadIdx.x;
  const int half = (lane >> 4) << 1;
  const bool mhi = (lane & 8) != 0;
  const int mrow = (lane & 7) * 8;

  // Basis fragments in registers. With m == n == lane&15:
  //   FA = A-frags of G  == B-frags of G^T  (masked row of M)
  //   FB = A-frags of G^T == B-frags of G   (masked row of M^T)
  const v2f mA = *(const v2f*)&c_M [mrow + half];
  const v2f mB = *(const v2f*)&c_M [mrow + 4 + half];
  const v2f tA = *(const v2f*)&c_MT[mrow + half];
  const v2f tB = *(const v2f*)&c_MT[mrow + 4 + half];
  const v2f z2 = {0.0f, 0.0f};
  const v2f FA[4] = { mhi ? z2 : mA, mhi ? z2 : mB, mhi ? mA : z2, mhi ? mB : z2 };
  const v2f FB[4] = { mhi ? z2 : tA, mhi ? z2 : tB, mhi ? tA : z2, mhi ? tB : z2 };

  // Per-lane quant rows (v = lane&7 fixed): qv[r] = Q[r][v]*0.4, iq = 1/qv
  const v4f qy0 = *(const v4f*)&c_Ystd[mrow];
  const v4f qy1 = *(const v4f*)&c_Ystd[mrow + 4];
  const v4f qc0 = *(const v4f*)&c_Cstd[mrow];
  const v4f qc1 = *(const v4f*)&c_Cstd[mrow + 4];
  float qvY[8], iqY[8], qvC[8], iqC[8];
#pragma unroll
  for (int r = 0; r < 8; ++r) {
    qvY[r] = (r < 4 ? qy0[r] : qy1[r - 4]) * QFACTOR;
    qvC[r] = (r < 4 ? qc0[r] : qc1[r - 4]) * QFACTOR;
    iqY[r] = __builtin_amdgcn_rcpf(qvY[r]);
    iqC[r] = __builtin_amdgcn_rcpf(qvC[r]);
  }
  const float av = ((lane & 7) == 0) ? INV_SQRT2 : 1.0f;

  // ---- load 16x16 RGB tile (float4), RGB->YCbCr ----
  const int tx = blockIdx.x, ty = blockIdx.y, bb = blockIdx.z;
  const size_t plane = (size_t)512 * 512;
  const size_t base  = (size_t)bb * 3 * plane;
  const int row0 = ty * 16, col0 = tx * 16;
  const int lsub = (lane & 3) * 4;       // col quad within tile
  const int lrb  = lane >> 2;            // 0..7
#pragma unroll
  for (int it = 0; it < 2; ++it) {
    const int lr = it * 8 + lrb;
    const size_t p = base + (size_t)(row0 + lr) * 512 + (col0 + lsub);
    v4f r  = *(const v4f*)&x[p]             * 255.0f;
    v4f g  = *(const v4f*)&x[p + plane]     * 255.0f;
    v4f bl = *(const v4f*)&x[p + 2 * plane] * 255.0f;
    v4f yv =  0.299f    * r + 0.587f    * g + 0.114f    * bl - 128.0f;
    v4f cb = -0.168736f * r - 0.331264f * g + 0.5f      * bl + 128.0f;
    v4f cr =  0.5f      * r - 0.418688f * g - 0.081312f * bl + 128.0f;
    *(v4f*)&Y [lr * 16 + lsub] = yv;
    *(v4f*)&T1[lr * 16 + lsub] = cb;
    *(v4f*)&T2[lr * 16 + lsub] = cr;
  }
  __syncthreads();

  // ---- 2x2 mean chroma downsample into blockdiag C = diag(Cb-128, Cr-128) ----
#pragma unroll
  for (int j = 0; j < 2; ++j) {
    const int idx = lane * 2 + j;          // 0..63
    const int r8 = idx >> 3, c8 = idx & 7;
    const int rr = r8 * 2, cc = c8 * 2;
    const v2f a0 = *(const v2f*)&T1[rr * 16 + cc];
    const v2f a1 = *(const v2f*)&T1[(rr + 1) * 16 + cc];
    const v2f b0 = *(const v2f*)&T2[rr * 16 + cc];
    const v2f b1 = *(const v2f*)&T2[(rr + 1) * 16 + cc];
    C[r8 * 16 + c8]             = 0.25f * (a0.x + a0.y + a1.x + a1.y) - 128.0f;
    C[(8 + r8) * 16 + (8 + c8)] = 0.25f * (b0.x + b0.y + b1.x + b1.y) - 128.0f;
    C[r8 * 16 + 8 + c8]         = 0.0f;    // off-diagonal zeros
    C[(8 + r8) * 16 + c8]       = 0.0f;
  }
  __syncthreads();

  process_plane(Y, qvY, iqY, FA, FB, T1, T2, lane, av);   // 4 Y blocks
  process_plane(C, qvC, iqC, FA, FB, T1, T2, lane, av);   // Cb + Cr blocks

  // ---- chroma upsample (nearest), YCbCr->RGB, clamp, /255, store float4 ----
#pragma unroll
  for (int it = 0; it < 2; ++it) {
    const int lr = it * 8 + lrb;
    const v4f yv = *(const v4f*)&Y[lr * 16 + lsub];
    const int hr = lr >> 1, hc = lsub >> 1;
    const v2f cb2 = *(const v2f*)&C[hr * 16 + hc];
    const v2f cr2 = *(const v2f*)&C[(8 + hr) * 16 + 8 + hc];
    v4f cb = { cb2.x, cb2.x, cb2.y, cb2.y };
    v4f cr = { cr2.x, cr2.x, cr2.y, cr2.y };
    cb -= 128.0f;
    cr -= 128.0f;
    v4f r  = yv + 1.402f * cr;
    v4f g  = yv - 0.344136f * cb - 0.714136f * cr;
    v4f bl = yv + 1.772f * cb;
#pragma unroll
    for (int c = 0; c < 4; ++c) {
      r[c]  = fminf(fmaxf(r[c],  0.0f), 255.0f) * (1.0f / 255.0f);
      g[c]  = fminf(fmaxf(g[c],  0.0f), 255.0f) * (1.0f / 255.0f);
      bl[c] = fminf(fmaxf(bl[c], 0.0f), 255.0f) * (1.0f / 255.0f);
    }
    const size_t p = base + (size_t)(row0 + lr) * 512 + (col0 + lsub);
    *(v4f*)&out[p]             = r;
    *(v4f*)&out[p + plane]     = g;
    *(v4f*)&out[p + 2 * plane] = bl;
  }
}

extern "C" void kernel_launch(void* const* d_in, const int* in_sizes, int n_in,
                              void* d_out, int out_size, void* d_ws, size_t ws_size,
                              hipStream_t stream) {
  (void)n_in; (void)d_ws; (void)ws_size; (void)out_size;
  const float* x   = (const float*)d_in[0];
  const float* src = (const float*)d_in[1];
  float* out = (float*)d_out;

  dim3 grid(512 / 16, 512 / 16, 16);
  diffjpeg_tile<<<grid, 32, 0, stream>>>(x, out);

  const size_t N = (size_t)in_sizes[1];
  hipMemcpyAsync(out + (size_t)in_sizes[0], src, N * sizeof(float),
                 hipMemcpyDeviceToDevice, stream);
}